// BlockAttnRes_20813411516558
// MI455X (gfx1250) — compile-verified
//
#include <hip/hip_runtime.h>

typedef float v2f __attribute__((ext_vector_type(2)));
typedef float v8f __attribute__((ext_vector_type(8)));
typedef int   v4i __attribute__((vector_size(16)));   // type expected by async-LDS builtin

#define N_SRC 8           // 7 blocks + partial
#define D 1024
#define BT 8192           // B*T = 4*2048
#define KC 64             // D-chunk staged per iteration (256B per row)
#define RSTRIDE 68        // LDS row stride in floats (bank padding)
#define TOK_PER_WG 16
#define WAVES 8
#define NCHUNK (D / KC)   // 16

#if __has_builtin(__builtin_amdgcn_global_load_async_to_lds_b128) && \
    __has_builtin(__builtin_amdgcn_s_wait_asynccnt)
#define USE_ASYNC_LDS 1
#else
#define USE_ASYNC_LDS 0
#endif

__global__ __launch_bounds__(256)
void block_attn_res_kernel(const float* __restrict__ blocks,   // (7,B,T,D)
                           const float* __restrict__ partial,  // (B,T,D)
                           const float* __restrict__ nw,       // (D,)
                           const float* __restrict__ w,        // (D,)
                           float* __restrict__ out)            // (B,T,D)
{
    __shared__ float wv[D];                          // w[d]*norm_weight[d]
    __shared__ float rowbuf[WAVES][2][16 * RSTRIDE]; // wave-private, double-buffered
    __shared__ float Ldot[WAVES][16];
    __shared__ float Llog[WAVES][16];
    __shared__ float Wgt[WAVES][2][N_SRC];

    const int tid  = threadIdx.x;
    const int lane = tid & 31;
    const int wave = tid >> 5;
    const int half = lane >> 4;    // 0: lanes 0-15, 1: lanes 16-31
    const int l15  = lane & 15;
    const size_t BTD = (size_t)BT * D;                 // stride between block sources
    const size_t gBase = (size_t)blockIdx.x * TOK_PER_WG;
    const size_t g = gBase + (size_t)wave * 2 + half;  // token of this half-wave

    // wv = w * norm_weight (shared, read-only after this barrier)
    #pragma unroll
    for (int k = 0; k < D / 256; ++k) {
        int d = tid + k * 256;
        wv[d] = w[d] * nw[d];
    }
    __syncthreads();

    // Per-lane global base pointer of row j (source j, this half-wave's token),
    // already offset to this lane's 16B column within a chunk.
    const float* gsrc[8];
    #pragma unroll
    for (int j = 0; j < 8; ++j)
        gsrc[j] = ((j < 7) ? (blocks + (size_t)j * BTD) : partial)
                  + g * D + (size_t)l15 * 4;

    // Stage one KC-chunk of 16 rows into rowbuf[wave][buf].
    auto stage = [&](int c, int buf) {
        #pragma unroll
        for (int j = 0; j < 8; ++j) {
            const float* gp = gsrc[j] + c * KC;
            float* lp = &rowbuf[wave][buf][(2 * j + half) * RSTRIDE + l15 * 4];
#if USE_ASYNC_LDS
            __builtin_amdgcn_global_load_async_to_lds_b128((v4i*)gp, (v4i*)lp, 0, 0);
#else
            *(float4*)lp = *(const float4*)gp;   // fallback: load+store per row
#endif
        }
    };

    // ---------------- Phase 1: logits via WMMA ----------------
    // Wave tile: 16 rows, row r = (source n = r>>1, token tk = r&1).
    // accG = V*V^T (diag = per-row sum of squares); accW = V*Bw (col0 = w-dot).
    v8f accG = {0.f,0.f,0.f,0.f,0.f,0.f,0.f,0.f};
    v8f accW = {0.f,0.f,0.f,0.f,0.f,0.f,0.f,0.f};

    stage(0, 0);
    for (int c = 0; c < NCHUNK; ++c) {
        const int cur = c & 1;
        if (c + 1 < NCHUNK) stage(c + 1, cur ^ 1);    // prefetch next chunk
#if USE_ASYNC_LDS
        asm volatile("" ::: "memory");
        if (c + 1 < NCHUNK) __builtin_amdgcn_s_wait_asynccnt(8);  // chunk c landed
        else                __builtin_amdgcn_s_wait_asynccnt(0);
        asm volatile("" ::: "memory");
#endif
        #pragma unroll
        for (int i = 0; i < KC / 4; ++i) {
            // A: lane holds row (lane&15), k-slots 2*half + {0,1}
            const v2f a  = *(const v2f*)&rowbuf[wave][cur][l15 * RSTRIDE + 4 * i + 2 * half];
            // B for w-dot: same k-slot mapping, broadcast across all 16 columns
            const v2f bw = *(const v2f*)&wv[c * KC + 4 * i + 2 * half];
            accG = __builtin_amdgcn_wmma_f32_16x16x4_f32(false, a, false, a,
                                                         (short)0, accG, false, false);
            accW = __builtin_amdgcn_wmma_f32_16x16x4_f32(false, a, false, bw,
                                                         (short)0, accW, false, false);
        }
    }

    // Dump accG tile into (reused) rowbuf to extract its diagonal.
    // C/D layout: VGPR r, lanes 0-15 -> (M=r, N=lane); lanes 16-31 -> (M=r+8, N=lane-16).
    {
        float* dump = &rowbuf[wave][0][0];
        const int mbase = half << 3;
        #pragma unroll
        for (int r = 0; r < 8; ++r)
            dump[(r + mbase) * 16 + l15] = accG[r];
        if (l15 == 0) {   // column 0 of accW: lane 0 -> M=0..7, lane 16 -> M=8..15
            #pragma unroll
            for (int r = 0; r < 8; ++r)
                Ldot[wave][r + mbase] = accW[r];
        }
    }
    __syncthreads();

    // logit = dot * rsqrt(mean(v^2) + eps), eps = f32 machine eps
    if (lane < 16) {
        const float ss = rowbuf[wave][0][lane * 16 + lane];
        const float dt = Ldot[wave][lane];
        Llog[wave][lane] = dt * rsqrtf(ss * (1.0f / (float)D) + 1.1920929e-07f);
    }
    __syncthreads();

    // Softmax over 8 sources, one lane per token (rows 2n+tk).
    if (lane < 2) {
        float lg[N_SRC];
        float mx = -3.402823466e+38f;
        #pragma unroll
        for (int n = 0; n < N_SRC; ++n) {
            lg[n] = Llog[wave][2 * n + lane];
            mx = fmaxf(mx, lg[n]);
        }
        float s = 0.f;
        #pragma unroll
        for (int n = 0; n < N_SRC; ++n) { lg[n] = __expf(lg[n] - mx); s += lg[n]; }
        const float inv = 1.0f / s;
        #pragma unroll
        for (int n = 0; n < N_SRC; ++n) Wgt[wave][lane][n] = lg[n] * inv;
    }
    __syncthreads();

    // ---------------- Phase 2: weighted combine (L2-resident re-read) ----------------
    float wn[N_SRC];
    #pragma unroll
    for (int n = 0; n < N_SRC; ++n) wn[n] = Wgt[wave][half][n];

    float* op = out + g * D;
    #pragma unroll 2
    for (int it = 0; it < NCHUNK; ++it) {
        const int d0 = l15 * 4 + it * KC;   // 256B coalesced runs per half-wave
        float4 acc = {0.f, 0.f, 0.f, 0.f};
        #pragma unroll
        for (int n = 0; n < N_SRC; ++n) {
            const float4 v = *(const float4*)(gsrc[n] - (size_t)l15 * 4 + d0);
            acc.x += wn[n] * v.x;
            acc.y += wn[n] * v.y;
            acc.z += wn[n] * v.z;
            acc.w += wn[n] * v.w;
        }
        *(float4*)(op + d0) = acc;
    }
}

extern "C" void kernel_launch(void* const* d_in, const int* in_sizes, int n_in,
                              void* d_out, int out_size, void* d_ws, size_t ws_size,
                              hipStream_t stream) {
    const float* blocks  = (const float*)d_in[0];  // (7,4,2048,1024)
    const float* partial = (const float*)d_in[1];  // (4,2048,1024)
    const float* nw      = (const float*)d_in[2];  // (1024,)
    const float* w       = (const float*)d_in[3];  // (1024,)
    float* out = (float*)d_out;                    // (4,2048,1024)

    const int grid = BT / TOK_PER_WG;  // 512 workgroups x 8 waves
    block_attn_res_kernel<<<grid, 256, 0, stream>>>(blocks, partial, nw, w, out);
}